// Router_8083128451229
// MI455X (gfx1250) — compile-verified
//
#include <hip/hip_runtime.h>
#include <hip/hip_bf16.h>

#define T_TOK   4096
#define DDIM    2048
#define NEXP    8
#define CAP     1280

typedef __attribute__((ext_vector_type(2))) float v2f;
typedef __attribute__((ext_vector_type(8))) float v8f;

// ---------------------------------------------------------------------------
// Kernel 1: gating logits via V_WMMA_F32_16X16X4_F32.
// B matrix columns 0..7 = w_g rows, columns 8..15 = w_noise rows, so one WMMA
// accumulation chain computes both einsums for a 16-token tile.
// Block = 256 threads = 8 waves; each wave owns a 16-token tile -> 128 tok/blk.
// ---------------------------------------------------------------------------
__global__ __launch_bounds__(256) void moe_gating_kernel(
    const float* __restrict__ x,       // (T, D)
    const float* __restrict__ noise,   // (T, 8)
    const float* __restrict__ w_g,     // (8, D)
    const float* __restrict__ w_noise, // (8, D)
    int*   __restrict__ eidx,          // (2, T) chosen expert per k
    float* __restrict__ probs)         // (2, T) router prob per k
{
    __shared__ float lds_logits[128][17];   // [token_local][16 cols], padded

    const int lane  = threadIdx.x & 31;
    const int wave  = threadIdx.x >> 5;
    const int tokBase = blockIdx.x * 128 + wave * 16;

    const int m     = lane & 15;          // token row within tile / weight row
    const int khalf = (lane >> 4) * 2;    // K sub-offset: 0 or 2

    const float* wrow = (m < 8) ? (w_g + (size_t)m * DDIM)
                                : (w_noise + (size_t)(m - 8) * DDIM);
    const float* xrow = x + (size_t)(tokBase + m) * DDIM;

    v8f acc = {};
    for (int k0 = 0; k0 < DDIM; k0 += 4) {
        v2f a = *(const v2f*)(xrow + k0 + khalf);   // A: 16x4 fp32 tile
        v2f b = *(const v2f*)(wrow + k0 + khalf);   // B: 4x16 fp32 tile (W^T)
        acc = __builtin_amdgcn_wmma_f32_16x16x4_f32(
            /*neg_a=*/false, a, /*neg_b=*/false, b,
            /*c_mod=*/(short)0, acc, /*reuse_a=*/false, /*reuse_b=*/false);
    }

    // C/D layout: acc[v] @ lane L -> row (v + (L>=16)*8), col (L&15)
    const int rowoff = (lane >> 4) * 8;
    const int col    = lane & 15;
#pragma unroll
    for (int v = 0; v < 8; ++v)
        lds_logits[wave * 16 + rowoff + v][col] = acc[v];

    __syncthreads();

    // Per-token post-processing: softplus noise, top-2, masked softmax.
    if (threadIdx.x < 128) {
        const int tl = threadIdx.x;
        const int t  = blockIdx.x * 128 + tl;
        float lg[NEXP];
#pragma unroll
        for (int e = 0; e < NEXP; ++e) {
            float g = lds_logits[tl][e];
            float n = lds_logits[tl][e + 8];
            float sp = (n > 20.f) ? n : log1pf(__expf(n));   // softplus
            lg[e] = g + sp * noise[(size_t)t * NEXP + e];
        }
        // top-2, first-max tie-break on lower index (matches jax.lax.top_k)
        int e0 = 0;
#pragma unroll
        for (int e = 1; e < NEXP; ++e) if (lg[e] > lg[e0]) e0 = e;
        int e1 = (e0 == 0) ? 1 : 0;
#pragma unroll
        for (int e = 0; e < NEXP; ++e) if (e != e0 && lg[e] > lg[e1]) e1 = e;

        float ex = __expf(lg[e1] - lg[e0]);   // lg[e1] <= lg[e0]
        float inv = 1.f / (1.f + ex);
        eidx[t]          = e0;
        eidx[T_TOK + t]  = e1;
        probs[t]         = inv;
        probs[T_TOK + t] = ex * inv;
    }
}

// ---------------------------------------------------------------------------
// Kernel 2: capacity ranks. k-major cumsum over (k*T, E) == per-expert
// prefix counts in token order, with k=1 offset by k=0 totals.
// Single workgroup, 1024 threads, 4 tokens/thread. Counters packed two
// 16-bit expert fields per u32 -> 4 u32 per k -> 8 scan arrays.
// ---------------------------------------------------------------------------
__global__ __launch_bounds__(1024) void moe_scan_kernel(
    const int* __restrict__ eidx,   // (2, T)
    int*       __restrict__ rnk)    // (2, T): rank in expert, or -1 if dropped
{
    __shared__ unsigned s[8][1024];
    const int tid = threadIdx.x;

    int e0l[4], e1l[4];
    unsigned loc[2][4] = {{0,0,0,0},{0,0,0,0}};
#pragma unroll
    for (int j = 0; j < 4; ++j) {
        int t = tid * 4 + j;
        int a0 = eidx[t];         e0l[j] = a0;
        int a1 = eidx[T_TOK + t]; e1l[j] = a1;
        loc[0][a0 >> 1] += 1u << ((a0 & 1) * 16);
        loc[1][a1 >> 1] += 1u << ((a1 & 1) * 16);
    }
#pragma unroll
    for (int q = 0; q < 8; ++q) s[q][tid] = loc[q >> 2][q & 3];
    __syncthreads();

    for (int off = 1; off < 1024; off <<= 1) {
        unsigned v[8];
#pragma unroll
        for (int q = 0; q < 8; ++q) v[q] = (tid >= off) ? s[q][tid - off] : 0u;
        __syncthreads();
#pragma unroll
        for (int q = 0; q < 8; ++q) s[q][tid] += v[q];
        __syncthreads();
    }

    // running counters = exclusive prefix; k=1 also offset by k=0 totals
    unsigned run[2][4];
#pragma unroll
    for (int q = 0; q < 8; ++q)
        run[q >> 2][q & 3] = s[q][tid] - loc[q >> 2][q & 3];
#pragma unroll
    for (int a = 0; a < 4; ++a) run[1][a] += s[a][1023];

#pragma unroll
    for (int j = 0; j < 4; ++j) {
        int t = tid * 4 + j;
        {
            int e = e0l[j]; int a = e >> 1; int sh = (e & 1) * 16;
            int r = (int)((run[0][a] >> sh) & 0xFFFFu);
            rnk[t] = (r < CAP) ? r : -1;
            run[0][a] += 1u << sh;
        }
        {
            int e = e1l[j]; int a = e >> 1; int sh = (e & 1) * 16;
            int r = (int)((run[1][a] >> sh) & 0xFFFFu);
            rnk[T_TOK + t] = (r < CAP) ? r : -1;
            run[1][a] += 1u << sh;
        }
    }
}

// ---------------------------------------------------------------------------
// Kernel 3a: zero entire output (outputs are overwhelmingly zero).
// ---------------------------------------------------------------------------
__global__ __launch_bounds__(256) void moe_zero_kernel(float4* __restrict__ p, long n4)
{
    long i = (long)blockIdx.x * blockDim.x + threadIdx.x;
    const long stride = (long)gridDim.x * blockDim.x;
    const float4 z = {0.f, 0.f, 0.f, 0.f};
    for (; i < n4; i += stride) p[i] = z;
}

// ---------------------------------------------------------------------------
// Kernel 3b: scatter router probs + mask into (T,E,cap).
// ---------------------------------------------------------------------------
__global__ __launch_bounds__(256) void moe_scatter_kernel(
    const int* __restrict__ eidx, const int* __restrict__ rnk,
    const float* __restrict__ probs,
    float* __restrict__ w_out, float* __restrict__ m_out)
{
    int t = blockIdx.x * blockDim.x + threadIdx.x;
    if (t >= T_TOK) return;
#pragma unroll
    for (int k = 0; k < 2; ++k) {
        int r = rnk[k * T_TOK + t];
        if (r >= 0) {
            int e = eidx[k * T_TOK + t];
            size_t off = ((size_t)t * NEXP + e) * CAP + r;
            w_out[off] = probs[k * T_TOK + t];
            m_out[off] = 1.0f;
        }
    }
}

// ---------------------------------------------------------------------------
// Kernel 3c: expert_batches[e, r, :] = x[t, :] for each kept assignment.
// One workgroup per (k, t) assignment: 8 KB row copy via float4.
// ---------------------------------------------------------------------------
__global__ __launch_bounds__(256) void moe_gather_kernel(
    const float* __restrict__ x, const int* __restrict__ eidx,
    const int* __restrict__ rnk, float* __restrict__ eb)
{
    const int b = blockIdx.x;          // 0 .. 2*T-1
    const int k = b >> 12;             // / 4096
    const int t = b & (T_TOK - 1);
    const int r = rnk[k * T_TOK + t];
    if (r < 0) return;
    const int e = eidx[k * T_TOK + t];
    const float4* src = (const float4*)(x + (size_t)t * DDIM);
    float4* dst = (float4*)(eb + ((size_t)e * CAP + r) * DDIM);
    for (int i = threadIdx.x; i < DDIM / 4; i += 256)
        dst[i] = src[i];
}

// ---------------------------------------------------------------------------
extern "C" void kernel_launch(void* const* d_in, const int* in_sizes, int n_in,
                              void* d_out, int out_size, void* d_ws, size_t ws_size,
                              hipStream_t stream) {
    (void)in_sizes; (void)n_in; (void)out_size; (void)ws_size;

    const float* x       = (const float*)d_in[0];
    const float* noise   = (const float*)d_in[1];
    const float* w_g     = (const float*)d_in[2];
    const float* w_noise = (const float*)d_in[3];

    // workspace layout
    int*   eidx  = (int*)d_ws;              // 2*T ints
    int*   rnk   = eidx + 2 * T_TOK;        // 2*T ints
    float* probs = (float*)(rnk + 2 * T_TOK); // 2*T floats

    // output layout: exp_weights | exp_mask_out | expert_batches
    const size_t wcnt = (size_t)T_TOK * NEXP * CAP;          // 41,943,040
    float* w_out = (float*)d_out;
    float* m_out = w_out + wcnt;
    float* eb    = m_out + wcnt;                             // E*CAP*D floats
    const long total_f = (long)(2 * wcnt + (size_t)NEXP * CAP * DDIM);

    // 1) zero the whole output (independent of gating; stream-serialized)
    moe_zero_kernel<<<4096, 256, 0, stream>>>((float4*)d_out, total_f / 4);

    // 2) WMMA gating + top-2 + softmax
    moe_gating_kernel<<<T_TOK / 128, 256, 0, stream>>>(x, noise, w_g, w_noise,
                                                       eidx, probs);

    // 3) capacity scan (single workgroup)
    moe_scan_kernel<<<1, 1024, 0, stream>>>(eidx, rnk);

    // 4) scatter weights / mask
    moe_scatter_kernel<<<(T_TOK + 255) / 256, 256, 0, stream>>>(eidx, rnk, probs,
                                                                w_out, m_out);

    // 5) gather x rows into expert_batches
    moe_gather_kernel<<<2 * T_TOK, 256, 0, stream>>>(x, eidx, rnk, eb);
}